// KernelRidge_18021682774449
// MI455X (gfx1250) — compile-verified
//
#include <hip/hip_runtime.h>
#include <hip/hip_bf16.h>

#define NN 4096
#define DD 256
#define TT 32
#define GAMMA (1.0f / 256.0f)
#define ALPHA 1.0f
#define CG_ITERS 200

typedef __attribute__((ext_vector_type(2))) float v2f;
typedef __attribute__((ext_vector_type(8))) float v8f;

// ---------------------------------------------------------------------------
// x2[i] = sum_j X[i][j]^2.  One wave32 per row, butterfly reduce.
// ---------------------------------------------------------------------------
__global__ __launch_bounds__(256) void rownorm_kernel(const float* __restrict__ X,
                                                      float* __restrict__ x2) {
    const int wave = threadIdx.x >> 5;
    const int lane = threadIdx.x & 31;
    const int row  = blockIdx.x * 8 + wave;
    float s = 0.0f;
    #pragma unroll
    for (int j = lane; j < DD; j += 32) {
        float v = X[row * DD + j];
        s += v * v;
    }
    #pragma unroll
    for (int off = 16; off > 0; off >>= 1) s += __shfl_xor(s, off, 32);
    if (lane == 0) x2[row] = s;
}

// ---------------------------------------------------------------------------
// A[i][j] = exp(-gamma * max(x2[i]+x2[j]-2*(X X^T)[i][j], 0)) + (i==j)*alpha
// One wave per 16x16 tile; G-tile via V_WMMA_F32_16X16X4_F32 over K=256.
// ---------------------------------------------------------------------------
__global__ __launch_bounds__(256) void build_A_kernel(const float* __restrict__ X,
                                                      const float* __restrict__ x2,
                                                      float* __restrict__ A) {
    const int wave = threadIdx.x >> 5;
    const int lane = threadIdx.x & 31;
    const int tile = blockIdx.x * 8 + wave;     // 256*256 tiles
    const int rt   = tile >> 8;
    const int ct   = tile & 255;
    const int grp  = lane >> 4;                 // 0 or 1
    const int lid  = lane & 15;

    const float* arow = X + (size_t)(rt * 16 + lid) * DD + 2 * grp;
    const float* brow = X + (size_t)(ct * 16 + lid) * DD + 2 * grp;

    v8f c = {};
    #pragma unroll 8
    for (int k = 0; k < DD; k += 4) {
        v2f a = *(const v2f*)(arow + k);
        v2f b = *(const v2f*)(brow + k);
        c = __builtin_amdgcn_wmma_f32_16x16x4_f32(false, a, false, b,
                                                  (short)0, c, false, false);
    }

    #pragma unroll
    for (int v = 0; v < 8; ++v) {
        const int i = rt * 16 + v + 8 * grp;
        const int j = ct * 16 + lid;
        float d2 = x2[i] + x2[j] - 2.0f * c[v];
        d2 = fmaxf(d2, 0.0f);
        float kij = __expf(-GAMMA * d2);
        if (i == j) kij += ALPHA;
        A[(size_t)i * NN + j] = kij;
    }
}

// ---------------------------------------------------------------------------
// AP = A(4096x4096) @ P(4096x32), P given transposed (Pt: 32 x 4096).
// One block (8 wave32) per 16-row stripe; each wave covers 512 of K with
// WMMA and computes BOTH 16-col tiles (A operand reused for 2 WMMAs, all
// operands single b64 loads).  Partials reduced via LDS; waves 0/1 emit the
// two column tiles.  A (64MB) streams from the 192MB L2.
// Block 0 also zeroes zero_t[0..31] (accumulator for the fused ||R||^2).
// ---------------------------------------------------------------------------
__global__ __launch_bounds__(256) void matmul_kernel(const float* __restrict__ A,
                                                     const float* __restrict__ Pt,
                                                     float* __restrict__ AP,
                                                     float* __restrict__ zero_t) {
    __shared__ float lds[2][8][256];
    const int wave = threadIdx.x >> 5;
    const int lane = threadIdx.x & 31;
    const int rt   = blockIdx.x;                // 256 row stripes
    const int grp  = lane >> 4;
    const int lid  = lane & 15;

    if (zero_t != nullptr && blockIdx.x == 0 && threadIdx.x < TT)
        zero_t[threadIdx.x] = 0.0f;

    const float* arow = A  + (size_t)(rt * 16 + lid) * NN + 2 * grp;
    const float* b0   = Pt + (size_t)(lid)      * NN + 2 * grp;   // cols 0..15
    const float* b1   = Pt + (size_t)(16 + lid) * NN + 2 * grp;   // cols 16..31

    v8f c0 = {}, c1 = {};
    const int k0 = wave * (NN / 8);             // 512 K per wave
    #pragma unroll 8
    for (int k = k0; k < k0 + NN / 8; k += 4) {
        v2f a  = *(const v2f*)(arow + k);
        v2f p0 = *(const v2f*)(b0 + k);
        v2f p1 = *(const v2f*)(b1 + k);
        c0 = __builtin_amdgcn_wmma_f32_16x16x4_f32(false, a, false, p0,
                                                   (short)0, c0, false, false);
        c1 = __builtin_amdgcn_wmma_f32_16x16x4_f32(false, a, false, p1,
                                                   (short)0, c1, false, false);
    }

    #pragma unroll
    for (int v = 0; v < 8; ++v) {
        lds[0][wave][v * 32 + lane] = c0[v];
        lds[1][wave][v * 32 + lane] = c1[v];
    }
    __syncthreads();

    if (wave < 2) {
        const int ct = wave;
        #pragma unroll
        for (int v = 0; v < 8; ++v) {
            float s = 0.0f;
            #pragma unroll
            for (int w = 0; w < 8; ++w) s += lds[ct][w][v * 32 + lane];
            const int i = rt * 16 + v + 8 * grp;
            const int j = ct * 16 + lid;
            AP[i * TT + j] = s;
        }
    }
}

// ---------------------------------------------------------------------------
// Per-column dot products: out[t] = sum_i U[i][t]*V[i][t].  One block/col.
// ---------------------------------------------------------------------------
__global__ __launch_bounds__(256) void dotcols_kernel(const float* __restrict__ U,
                                                      const float* __restrict__ V,
                                                      float* __restrict__ out) {
    __shared__ float sm[256];
    const int t = blockIdx.x;
    float s = 0.0f;
    for (int i = threadIdx.x; i < NN; i += 256)
        s += U[i * TT + t] * V[i * TT + t];
    sm[threadIdx.x] = s;
    __syncthreads();
    for (int off = 128; off > 0; off >>= 1) {
        if (threadIdx.x < off) sm[threadIdx.x] += sm[threadIdx.x + off];
        __syncthreads();
    }
    if (threadIdx.x == 0) out[t] = sm[0];
}

// ---------------------------------------------------------------------------
// LDS transpose helper: block covers rows [i0, i0+8) x 32 cols, writes Pt.
// Thread tid holds val for (i0 + (tid>>5), tid&31).
// ---------------------------------------------------------------------------
__device__ __forceinline__ void store_transposed(float* __restrict__ Pt, int i0,
                                                 float val, float* sm) {
    const int tid = threadIdx.x;
    sm[tid] = val;                      // sm[il*32 + t]
    __syncthreads();
    const int t2 = tid >> 3;            // 0..31
    const int il = tid & 7;             // 0..7
    Pt[(size_t)t2 * NN + i0 + il] = sm[il * 32 + t2];
}

__global__ __launch_bounds__(256) void cg_init_kernel(const float* __restrict__ y,
                                                      float* __restrict__ R,
                                                      float* __restrict__ P,
                                                      float* __restrict__ Pt,
                                                      float* __restrict__ dual) {
    __shared__ float sm[256];
    const int idx = blockIdx.x * 256 + threadIdx.x;
    const float v = y[idx];
    R[idx] = v;
    P[idx] = v;
    dual[idx] = 0.0f;
    store_transposed(Pt, blockIdx.x * 8, v, sm);
}

// dual += alpha*P ; R -= alpha*AP ; rrn[t] += sum(newR^2) (block-reduced
// through LDS, then one global f32 atomic per column per block).
__global__ __launch_bounds__(256) void cg_update1_kernel(const float* __restrict__ P,
                                                         const float* __restrict__ AP,
                                                         const float* __restrict__ rr,
                                                         const float* __restrict__ pap,
                                                         float* __restrict__ dual,
                                                         float* __restrict__ R,
                                                         float* __restrict__ rrn) {
    __shared__ float sm[256];
    const int idx = blockIdx.x * 256 + threadIdx.x;
    const int t = idx & (TT - 1);
    const float d = pap[t];
    const float a = (d != 0.0f) ? rr[t] / d : 0.0f;
    dual[idx] += a * P[idx];
    const float nr = R[idx] - a * AP[idx];
    R[idx] = nr;

    sm[threadIdx.x] = nr * nr;          // sm[il*32 + t]
    __syncthreads();
    if (threadIdx.x < 128) sm[threadIdx.x] += sm[threadIdx.x + 128];
    __syncthreads();
    if (threadIdx.x < 64)  sm[threadIdx.x] += sm[threadIdx.x + 64];
    __syncthreads();
    if (threadIdx.x < 32) {
        const float s = sm[threadIdx.x] + sm[threadIdx.x + 32];
        atomicAdd(&rrn[threadIdx.x], s);
    }
}

// P = R + beta*P (beta[t] = rrn[t]/rr[t]); also refresh Pt via LDS transpose.
__global__ __launch_bounds__(256) void cg_update2_kernel(const float* __restrict__ R,
                                                         const float* __restrict__ rr,
                                                         const float* __restrict__ rrn,
                                                         float* __restrict__ P,
                                                         float* __restrict__ Pt) {
    __shared__ float sm[256];
    const int idx = blockIdx.x * 256 + threadIdx.x;
    const int t = idx & (TT - 1);
    const float d = rr[t];
    const float b = (d != 0.0f) ? rrn[t] / d : 0.0f;
    const float val = R[idx] + b * P[idx];
    P[idx] = val;
    store_transposed(Pt, blockIdx.x * 8, val, sm);
}

// Pt = transpose(dual) for the final matvec.
__global__ __launch_bounds__(256) void transpose_kernel(const float* __restrict__ V,
                                                        float* __restrict__ Pt) {
    __shared__ float sm[256];
    const int idx = blockIdx.x * 256 + threadIdx.x;
    store_transposed(Pt, blockIdx.x * 8, V[idx], sm);
}

// out = AP - dual   (y_pred = A*dual - dual = K*dual)
__global__ __launch_bounds__(256) void final_kernel(const float* __restrict__ AP,
                                                    const float* __restrict__ dual,
                                                    float* __restrict__ out) {
    const int idx = blockIdx.x * 256 + threadIdx.x;
    out[idx] = AP[idx] - dual[idx];
}

extern "C" void kernel_launch(void* const* d_in, const int* in_sizes, int n_in,
                              void* d_out, int out_size, void* d_ws, size_t ws_size,
                              hipStream_t stream) {
    const float* X = (const float*)d_in[0];   // [N, D]
    const float* y = (const float*)d_in[1];   // [N, T]
    float* out = (float*)d_out;               // [N, T]

    // workspace layout
    float* A    = (float*)d_ws;               // N*N   (64 MB, fits in L2)
    float* x2   = A    + (size_t)NN * NN;     // N
    float* R    = x2   + NN;                  // N*T
    float* P    = R    + (size_t)NN * TT;     // N*T
    float* Pt   = P    + (size_t)NN * TT;     // T*N  (transposed search dirs)
    float* AP   = Pt   + (size_t)NN * TT;     // N*T
    float* dual = AP   + (size_t)NN * TT;     // N*T
    float* s0   = dual + (size_t)NN * TT;     // T  (ping-pong ||R||^2)
    float* s1   = s0 + TT;                    // T
    float* pap  = s1 + TT;                    // T

    const int elemBlocks = (NN * TT) / 256;   // 512

    // 1) row norms + kernel matrix A = K + alpha*I (WMMA f32 GEMM + exp)
    rownorm_kernel<<<NN / 8, 256, 0, stream>>>(X, x2);
    build_A_kernel<<<(256 * 256) / 8, 256, 0, stream>>>(X, x2, A);

    // 2) blocked CG: solve A dual = y (32 RHS, per-column scalars)
    cg_init_kernel<<<elemBlocks, 256, 0, stream>>>(y, R, P, Pt, dual);
    dotcols_kernel<<<TT, 256, 0, stream>>>(R, R, s0);

    float* rrbuf[2] = { s0, s1 };
    for (int it = 0; it < CG_ITERS; ++it) {
        float* cur = rrbuf[it & 1];           // rr      = ||R||^2 (old)
        float* nxt = rrbuf[(it + 1) & 1];     // rr_new  accumulator
        matmul_kernel<<<256, 256, 0, stream>>>(A, Pt, AP, nxt);   // also zeroes nxt
        dotcols_kernel<<<TT, 256, 0, stream>>>(P, AP, pap);
        cg_update1_kernel<<<elemBlocks, 256, 0, stream>>>(P, AP, cur, pap,
                                                          dual, R, nxt);
        cg_update2_kernel<<<elemBlocks, 256, 0, stream>>>(R, cur, nxt, P, Pt);
    }

    // 3) y_pred = K @ dual = A @ dual - dual
    transpose_kernel<<<elemBlocks, 256, 0, stream>>>(dual, Pt);
    matmul_kernel<<<256, 256, 0, stream>>>(A, Pt, AP, nullptr);
    final_kernel<<<elemBlocks, 256, 0, stream>>>(AP, dual, out);
}